// AttentionModel_6468220748059
// MI455X (gfx1250) — compile-verified
//
#include <hip/hip_runtime.h>
#include <math.h>

// Problem constants (match reference)
#define B_   32
#define L_   4100
#define S_   1024
#define E_   16
#define H_   4
#define DH_  4
#define NC_  10
#define EPS_ 1e-5f

typedef float v2f __attribute__((ext_vector_type(2)));
typedef float v8f __attribute__((ext_vector_type(8)));

// D = A(16x4 f32) * B(4x16 f32) + C(16x16 f32), wave32, one tile per wave.
__device__ __forceinline__ v8f wmma4(v2f a, v2f b, v8f c) {
    return __builtin_amdgcn_wmma_f32_16x16x4_f32(
        /*neg_a=*/false, a, /*neg_b=*/false, b,
        /*c_mod=*/(short)0, c, /*reuse_a=*/false, /*reuse_b=*/false);
}

// ---------------------------------------------------------------- PE table
__global__ void pe_kernel(float* __restrict__ pe) {
    int idx = blockIdx.x * blockDim.x + threadIdx.x;
    if (idx >= S_ * E_) return;
    int s = idx / E_, e = idx % E_;
    float j   = (float)(e & ~1);                       // 2*(e/2)
    float div = __expf(j * (-logf(10000.0f) / (float)E_));
    float ang = (float)s * div * ((float)E_ / (float)S_);
    pe[idx] = (e & 1) ? __cosf(ang) : __sinf(ang);
}

// ------------------------------------------- conv1: 1->8, k=8, stride=4, valid
__global__ void conv1_kernel(const float* __restrict__ x, const float* __restrict__ w,
                             const float* __restrict__ bias, const float* __restrict__ g,
                             const float* __restrict__ bb, float* __restrict__ out) {
    int idx = blockIdx.x * blockDim.x + threadIdx.x;
    if (idx >= B_ * 8 * S_) return;
    int s = idx % S_, c = (idx / S_) % 8, b = idx / (8 * S_);
    const float* xp = x + (size_t)b * L_ + 4 * s;
    const float* wp = w + c * 8;
    float acc = bias[c];
#pragma unroll
    for (int k = 0; k < 8; ++k) acc += xp[k] * wp[k];
    float bn = acc * rsqrtf(1.0f + EPS_) * g[c] + bb[c];
    out[idx] = fmaxf(bn, 0.0f);
}

// --------------------------------------- conv2: 8->32, k=8, same pad (3,4)
__global__ void conv2_kernel(const float* __restrict__ h1, const float* __restrict__ w,
                             const float* __restrict__ bias, const float* __restrict__ g,
                             const float* __restrict__ bb, float* __restrict__ out) {
    int idx = blockIdx.x * blockDim.x + threadIdx.x;
    if (idx >= B_ * 32 * S_) return;
    int s = idx % S_, o = (idx / S_) % 32, b = idx / (32 * S_);
    float acc = bias[o];
#pragma unroll
    for (int i = 0; i < 8; ++i) {
        const float* hp = h1 + ((size_t)b * 8 + i) * S_;
        const float* wp = w + ((size_t)o * 8 + i) * 8;
#pragma unroll
        for (int k = 0; k < 8; ++k) {
            int t = s - 3 + k;
            if (t >= 0 && t < S_) acc += hp[t] * wp[k];
        }
    }
    float bn = acc * rsqrtf(1.0f + EPS_) * g[o] + bb[o];
    out[idx] = fmaxf(bn, 0.0f);
}

// --------------- depthwise-grouped conv: 32->16, k=32, groups=16, pad (15,16)
__global__ void convdw_kernel(const float* __restrict__ h2, const float* __restrict__ w,
                              float* __restrict__ out) {
    int idx = blockIdx.x * blockDim.x + threadIdx.x;
    if (idx >= B_ * 16 * S_) return;
    int s = idx % S_, gr = (idx / S_) % 16, b = idx / (16 * S_);
    float acc = 0.0f;
#pragma unroll
    for (int ci = 0; ci < 2; ++ci) {
        const float* hp = h2 + ((size_t)b * 32 + 2 * gr + ci) * S_;
        const float* wp = w + ((size_t)gr * 2 + ci) * 32;
        for (int k = 0; k < 32; ++k) {
            int t = s - 15 + k;
            if (t >= 0 && t < S_) acc += hp[t] * wp[k];
        }
    }
    out[idx] = acc;
}

// ----- pointwise 16->16 + BN + ReLU, writes transposed x_src [B][S][16]
__global__ void convpw_kernel(const float* __restrict__ dw, const float* __restrict__ w,
                              const float* __restrict__ g, const float* __restrict__ bb,
                              float* __restrict__ xsrc) {
    int idx = blockIdx.x * blockDim.x + threadIdx.x;  // (b,s,o), o fastest
    if (idx >= B_ * S_ * E_) return;
    int o = idx % E_, s = (idx / E_) % S_, b = idx / (E_ * S_);
    float acc = 0.0f;
#pragma unroll
    for (int c = 0; c < 16; ++c) acc += dw[((size_t)b * 16 + c) * S_ + s] * w[o * 16 + c];
    float bn = acc * rsqrtf(1.0f + EPS_) * g[o] + bb[o];
    xsrc[idx] = fmaxf(bn, 0.0f);
}

// ---------------- QKV projection: one wave per (b, s-tile); 12 WMMAs / wave.
// A = (in+pe) tile [16x16] split into 4 K-chunks of 4; B[k][n] = W[n][k].
// A layout (f32 16x4): lane l -> m = l%16, k = (l/16)*2 + p (p = vgpr 0/1).
__global__ __launch_bounds__(128) void qkv_kernel(
    const float* __restrict__ in, const float* __restrict__ pe,
    const float* __restrict__ wq, const float* __restrict__ wk,
    const float* __restrict__ wv,
    float* __restrict__ Q, float* __restrict__ K, float* __restrict__ V) {
    int wave = (blockIdx.x * blockDim.x + threadIdx.x) >> 5;  // 0..B*S/16-1
    int lane = threadIdx.x & 31;
    int g = lane >> 4, n = lane & 15;
    int stile = wave & (S_ / 16 - 1);
    int b = wave >> 6;
    int s0 = stile * 16;

    const float* rowin = in + ((size_t)b * S_ + s0 + n) * E_;
    const float* rowpe = pe + (size_t)(s0 + n) * E_;
    v2f a[4];
#pragma unroll
    for (int c = 0; c < 4; ++c) {
        int k0 = 4 * c + 2 * g;
        a[c][0] = rowin[k0] + rowpe[k0];
        a[c][1] = rowin[k0 + 1] + rowpe[k0 + 1];
    }
    const float* Ws[3] = {wq, wk, wv};
    float* Os[3] = {Q, K, V};
#pragma unroll
    for (int m = 0; m < 3; ++m) {
        const float* W = Ws[m];
        v2f bm[4];
#pragma unroll
        for (int c = 0; c < 4; ++c) {
            int k0 = 4 * c + 2 * g;
            bm[c][0] = W[n * E_ + k0];      // B[k][n] = W[n][k]
            bm[c][1] = W[n * E_ + k0 + 1];
        }
        v8f acc = {};
#pragma unroll
        for (int c = 0; c < 4; ++c) acc = wmma4(a[c], bm[c], acc);
        float* O = Os[m] + ((size_t)b * S_ + s0) * E_;
#pragma unroll
        for (int i = 0; i < 8; ++i) O[(i + 8 * g) * E_ + n] = acc[i];
    }
}

// ------- flash attention w/ multiplicative distance weight, one wave per
// (b, h, s-tile). Score tile computed TRANSPOSED: D[t][s] = K(16x4) x Q^T(4x16),
// so each lane (g = lane/16, n = lane%16) owns query s = s0+n and t values
// {t0 + i + 8g}. Row softmax = per-lane reduce over 8 regs + one shfl_xor(16).
__global__ __launch_bounds__(128) void attn_kernel(
    const float* __restrict__ Q, const float* __restrict__ K,
    const float* __restrict__ V, float* __restrict__ out) {
    int wave = (blockIdx.x * blockDim.x + threadIdx.x) >> 5;  // 0..B*H*S/16-1
    int lane = threadIdx.x & 31;
    int g = lane >> 4, n = lane & 15;
    int stile = wave & (S_ / 16 - 1);
    int h = (wave >> 6) & (H_ - 1);
    int b = wave >> 8;
    int s0 = stile * 16;
    const float scale = 0.25f;  // E^-0.5

    // Loop-invariant B operand: B[k][n] = Q[s0+n][h*4+k], k = p + 2g.
    v2f qb;
    {
        const float* qp = Q + ((size_t)b * S_ + s0 + n) * E_ + h * DH_ + 2 * g;
        qb[0] = qp[0];
        qb[1] = qp[1];
    }

    float mrun = -3.0e38f, lrun = 0.0f;
    float acc[4] = {0.0f, 0.0f, 0.0f, 0.0f};

    for (int tt = 0; tt < S_ / 16; ++tt) {
        int t0 = tt * 16;
        // A operand: K tile, A[m][k] = K[t0+m][h*4+k], m = n, k = p + 2g.
        v2f ka;
        const float* kp = K + ((size_t)b * S_ + t0 + n) * E_ + h * DH_ + 2 * g;
        ka[0] = kp[0];
        ka[1] = kp[1];
        v8f zero = {};
        v8f sc = wmma4(ka, qb, zero);  // sc[i] = score(t = t0+i+8g, s = s0+n)

        float sv[8];
        float lm = -3.0e38f;
#pragma unroll
        for (int i = 0; i < 8; ++i) {
            int t = t0 + i + 8 * g;
            float w = fabsf((float)(t - (s0 + n))) * (1.0f / (float)S_);
            sv[i] = sc[i] * scale * w;
            lm = fmaxf(lm, sv[i]);
        }
        lm = fmaxf(lm, __shfl_xor(lm, 16));      // merge the two t-halves
        float mnew  = fmaxf(mrun, lm);
        float alpha = __expf(mrun - mnew);
        mrun = mnew;

        float p[8];
        float rs = 0.0f;
#pragma unroll
        for (int i = 0; i < 8; ++i) {
            p[i] = __expf(sv[i] - mnew);
            rs += p[i];
        }
        rs += __shfl_xor(rs, 16);
        lrun = lrun * alpha + rs;

#pragma unroll
        for (int d = 0; d < 4; ++d) acc[d] *= alpha;
        const float* vbase = V + ((size_t)b * S_ + t0 + 8 * g) * E_ + h * DH_;
#pragma unroll
        for (int i = 0; i < 8; ++i) {
            float4 vv = *(const float4*)(vbase + (size_t)i * E_);  // uniform in half-wave
            acc[0] += p[i] * vv.x;
            acc[1] += p[i] * vv.y;
            acc[2] += p[i] * vv.z;
            acc[3] += p[i] * vv.w;
        }
    }

    float inv = 1.0f / lrun;
    float o[4];
#pragma unroll
    for (int d = 0; d < 4; ++d) {
        float a = acc[d] + __shfl_xor(acc[d], 16);  // combine half-wave partials
        o[d] = a * inv;
    }
    if (g == 0) {
        float4 res = make_float4(o[0], o[1], o[2], o[3]);
        *(float4*)(out + ((size_t)b * S_ + s0 + n) * E_ + h * DH_) = res;
    }
}

// ------------- fused double LayerNorm: z = LN2( LN1(a) + resid )
__global__ void ln2_kernel(const float* __restrict__ ain, const float* __restrict__ resid,
                           const float* __restrict__ g1, const float* __restrict__ b1,
                           const float* __restrict__ g2, const float* __restrict__ b2,
                           float* __restrict__ out) {
    int idx = blockIdx.x * blockDim.x + threadIdx.x;  // (b,s)
    if (idx >= B_ * S_) return;
    const float* a = ain + (size_t)idx * E_;
    float y[E_];
    float mu = 0.0f;
#pragma unroll
    for (int e = 0; e < E_; ++e) { y[e] = a[e]; mu += y[e]; }
    mu *= (1.0f / E_);
    float var = 0.0f;
#pragma unroll
    for (int e = 0; e < E_; ++e) { float d = y[e] - mu; var += d * d; }
    var *= (1.0f / E_);
    float r = rsqrtf(var + EPS_);
#pragma unroll
    for (int e = 0; e < E_; ++e) {
        float t = (y[e] - mu) * r * g1[e] + b1[e];
        y[e] = t + (resid ? resid[(size_t)idx * E_ + e] : 0.0f);
    }
    float mu2 = 0.0f;
#pragma unroll
    for (int e = 0; e < E_; ++e) mu2 += y[e];
    mu2 *= (1.0f / E_);
    float var2 = 0.0f;
#pragma unroll
    for (int e = 0; e < E_; ++e) { float d = y[e] - mu2; var2 += d * d; }
    var2 *= (1.0f / E_);
    float r2 = rsqrtf(var2 + EPS_);
#pragma unroll
    for (int e = 0; e < E_; ++e)
        out[(size_t)idx * E_ + e] = (y[e] - mu2) * r2 * g2[e] + b2[e];
}

// ------------- mean-pool over S then 16->10 head
__global__ __launch_bounds__(256) void head_kernel(const float* __restrict__ att,
                                                   const float* __restrict__ ow,
                                                   const float* __restrict__ ob,
                                                   float* __restrict__ out) {
    __shared__ float sums[E_];
    int b = blockIdx.x;
    int t = threadIdx.x;
    if (t < E_) sums[t] = 0.0f;
    __syncthreads();
    float part[E_];
#pragma unroll
    for (int e = 0; e < E_; ++e) part[e] = 0.0f;
    for (int s = t; s < S_; s += 256) {
        const float* r = att + ((size_t)b * S_ + s) * E_;
#pragma unroll
        for (int e = 0; e < E_; ++e) part[e] += r[e];
    }
#pragma unroll
    for (int e = 0; e < E_; ++e) atomicAdd(&sums[e], part[e]);
    __syncthreads();
    if (t < NC_) {
        float acc = ob[t];
#pragma unroll
        for (int e = 0; e < E_; ++e) acc += (sums[e] * (1.0f / S_)) * ow[t * E_ + e];
        out[b * NC_ + t] = acc;
    }
}

extern "C" void kernel_launch(void* const* d_in, const int* in_sizes, int n_in,
                              void* d_out, int out_size, void* d_ws, size_t ws_size,
                              hipStream_t stream) {
    (void)in_sizes; (void)n_in; (void)out_size; (void)ws_size;
    const float* x       = (const float*)d_in[0];
    const float* patch_w = (const float*)d_in[1];
    const float* patch_b = (const float*)d_in[2];
    const float* bn1_g   = (const float*)d_in[3];
    const float* bn1_b   = (const float*)d_in[4];
    const float* emb_w   = (const float*)d_in[5];
    const float* emb_b   = (const float*)d_in[6];
    const float* bn2_g   = (const float*)d_in[7];
    const float* bn2_b   = (const float*)d_in[8];
    const float* dw_w    = (const float*)d_in[9];
    const float* pw_w    = (const float*)d_in[10];
    const float* bn3_g   = (const float*)d_in[11];
    const float* bn3_b   = (const float*)d_in[12];
    const float* q1_w    = (const float*)d_in[13];
    const float* k1_w    = (const float*)d_in[14];
    const float* v1_w    = (const float*)d_in[15];
    const float* lna1_g  = (const float*)d_in[16];
    const float* lna1_b  = (const float*)d_in[17];
    const float* ln1_g   = (const float*)d_in[18];
    const float* ln1_b   = (const float*)d_in[19];
    const float* q2_w    = (const float*)d_in[20];
    const float* k2_w    = (const float*)d_in[21];
    const float* v2_w    = (const float*)d_in[22];
    const float* lna2_g  = (const float*)d_in[23];
    const float* lna2_b  = (const float*)d_in[24];
    const float* ln2_g   = (const float*)d_in[25];
    const float* ln2_b   = (const float*)d_in[26];
    const float* out_w   = (const float*)d_in[27];
    const float* out_b   = (const float*)d_in[28];

    float* ws   = (float*)d_ws;
    float* pe   = ws;                       // S*E        = 16384
    float* xsrc = pe   + (size_t)S_ * E_;   // B*S*E      = 524288
    float* z1   = xsrc + (size_t)B_ * S_ * E_;
    float* z2   = z1   + (size_t)B_ * S_ * E_;
    float* Qb   = z2   + (size_t)B_ * S_ * E_;
    float* Kb   = Qb   + (size_t)B_ * S_ * E_;
    float* Vb   = Kb   + (size_t)B_ * S_ * E_;
    float* attn = Vb   + (size_t)B_ * S_ * E_;
    float* h1   = attn + (size_t)B_ * S_ * E_;   // B*8*S
    float* h2   = h1   + (size_t)B_ * 8 * S_;    // B*32*S
    float* dwb  = h2   + (size_t)B_ * 32 * S_;   // B*16*S

    const int qkv_blocks  = (B_ * (S_ / 16) * 32) / 128;        // 512
    const int attn_blocks = (B_ * H_ * (S_ / 16) * 32) / 128;   // 2048

    pe_kernel<<<(S_ * E_ + 255) / 256, 256, 0, stream>>>(pe);
    conv1_kernel<<<(B_ * 8 * S_ + 255) / 256, 256, 0, stream>>>(x, patch_w, patch_b, bn1_g, bn1_b, h1);
    conv2_kernel<<<(B_ * 32 * S_ + 255) / 256, 256, 0, stream>>>(h1, emb_w, emb_b, bn2_g, bn2_b, h2);
    convdw_kernel<<<(B_ * 16 * S_ + 255) / 256, 256, 0, stream>>>(h2, dw_w, dwb);
    convpw_kernel<<<(B_ * S_ * E_ + 255) / 256, 256, 0, stream>>>(dwb, pw_w, bn3_g, bn3_b, xsrc);

    // Block 1
    qkv_kernel<<<qkv_blocks, 128, 0, stream>>>(xsrc, pe, q1_w, k1_w, v1_w, Qb, Kb, Vb);
    attn_kernel<<<attn_blocks, 128, 0, stream>>>(Qb, Kb, Vb, attn);
    ln2_kernel<<<(B_ * S_ + 255) / 256, 256, 0, stream>>>(attn, xsrc, lna1_g, lna1_b, ln1_g, ln1_b, z1);

    // Block 2
    qkv_kernel<<<qkv_blocks, 128, 0, stream>>>(z1, pe, q2_w, k2_w, v2_w, Qb, Kb, Vb);
    attn_kernel<<<attn_blocks, 128, 0, stream>>>(Qb, Kb, Vb, attn);
    ln2_kernel<<<(B_ * S_ + 255) / 256, 256, 0, stream>>>(attn, nullptr, lna2_g, lna2_b, ln2_g, ln2_b, z2);

    head_kernel<<<B_, 256, 0, stream>>>(z2, out_w, out_b, (float*)d_out);
}